// PraxisExpert_58128087384380
// MI455X (gfx1250) — compile-verified
//
#include <hip/hip_runtime.h>

typedef __attribute__((ext_vector_type(2))) float v2f;
typedef __attribute__((ext_vector_type(8))) float v8f;

constexpr int Bdim = 4, S = 2048, D = 512, Hdim = 2048, E = 8, KSEL = 2;
constexpr int T  = Bdim * S;       // 8192 tokens
constexpr int TK = T * KSEL;       // 16384 (token, expert) pairs
constexpr int TILE_M = 16;
constexpr int MAX_TILES = TK / TILE_M + E;  // 1032 worst-case tiles
constexpr int HC = 256;                      // H chunk held in LDS
constexpr int NCHUNK = Hdim / HC;            // 8

// workspace layout in ints
#define WS_COUNTS 0
#define WS_OFFS   8
#define WS_CUR    17
#define WS_PAIRS  32

__global__ void k_zero(int* ws) {
    int i = threadIdx.x;
    if (i < E) { ws[WS_COUNTS + i] = 0; ws[WS_CUR + i] = 0; }
}

__global__ void k_count(const int* __restrict__ idx, int* __restrict__ ws) {
    int p = blockIdx.x * blockDim.x + threadIdx.x;
    if (p < TK) atomicAdd(&ws[WS_COUNTS + idx[p]], 1);
}

__global__ void k_scan(int* ws) {
    if (threadIdx.x == 0) {
        int acc = 0;
        for (int e = 0; e < E; ++e) {
            ws[WS_OFFS + e] = acc;
            ws[WS_CUR + e]  = acc;
            acc += ws[WS_COUNTS + e];
        }
        ws[WS_OFFS + E] = acc;
    }
}

__global__ void k_fill(const int* __restrict__ idx, int* __restrict__ ws) {
    int p = blockIdx.x * blockDim.x + threadIdx.x;
    if (p < TK) {
        int e = idx[p];
        int pos = atomicAdd(&ws[WS_CUR + e], 1);
        ws[WS_PAIRS + pos] = p;
    }
}

// Fused per-tile MoE FFN: out_row = relu(x_row @ W1[e] + b1[e]) @ W2[e] + b2[e]
__global__ __launch_bounds__(256, 1) void k_moe(
    const float* __restrict__ x,  const float* __restrict__ W1,
    const float* __restrict__ b1, const float* __restrict__ W2,
    const float* __restrict__ b2, const int* __restrict__ ws,
    float* __restrict__ out)
{
    __shared__ float Xs[TILE_M][D + 4];    // +4 pad: conflict-free A reads
    __shared__ float Hc[TILE_M][HC + 4];
    __shared__ int   sPair[TILE_M];

    const int tid  = threadIdx.x;
    const int lane = tid & 31;
    const int wv   = tid >> 5;
    const int ln   = lane & 15;
    const int khalf = (lane >> 4) << 1;    // 0 for lanes 0-15, 2 for lanes 16-31

    // map blockIdx -> (expert, tile-in-expert, pair-list offset)
    int bid = blockIdx.x;
    int e = -1, tIdx = 0, off = 0;
    {
        int accT = 0, accC = 0;
        for (int i = 0; i < E; ++i) {
            int c  = ws[WS_COUNTS + i];
            int nt = (c + TILE_M - 1) / TILE_M;
            if (e < 0 && bid < accT + nt) { e = i; tIdx = bid - accT; off = accC; }
            accT += nt; accC += c;
        }
    }
    if (e < 0) return;
    const int cnt  = ws[WS_COUNTS + e];
    const int rows = min(TILE_M, cnt - tIdx * TILE_M);

    if (tid < TILE_M)
        sPair[tid] = (tid < rows) ? ws[WS_PAIRS + off + tIdx * TILE_M + tid] : -1;
    __syncthreads();

    // stage X tile (16 x 512) into LDS, float4 vectorized; 2048 float4 / 256 thr
    #pragma unroll
    for (int it = 0; it < 8; ++it) {
        int idx4 = it * 256 + tid;
        int m  = idx4 >> 7;      // /128 float4 per row
        int c4 = idx4 & 127;
        float4 v = make_float4(0.f, 0.f, 0.f, 0.f);
        int p = sPair[m];
        if (p >= 0) {
            int tok = p >> 1;    // KSEL == 2
            v = *(const float4*)(x + (size_t)tok * D + c4 * 4);
        }
        *(float4*)(&Xs[m][c4 * 4]) = v;
    }
    __syncthreads();

    const float* W1e = W1 + (size_t)e * D * Hdim;
    const float* W2e = W2 + (size_t)e * Hdim * D;
    const float* b1e = b1 + (size_t)e * Hdim;
    const float* b2e = b2 + (size_t)e * D;

    v8f accY[4];
    #pragma unroll
    for (int i = 0; i < 4; ++i) accY[i] = {};

    const int mbase = (lane >> 4) * 8;     // C/D row base for this half-wave

    for (int chunk = 0; chunk < NCHUNK; ++chunk) {
        // ---- stage 1: Hc[:, wv*32 .. +32) = relu(X @ W1[:, chunk cols] + b1)
        const int n0 = chunk * HC + wv * 32;
        v8f h0 = {}, h1 = {};
        const float* w1p  = W1e + khalf * Hdim + n0 + ln;
        const float* xrow = &Xs[ln][khalf];
        #pragma unroll 4
        for (int k = 0; k < D; k += 4) {
            v2f a = *(const v2f*)(xrow + k);       // A: M=ln, K=k+khalf..+1
            v2f bA, bB;
            bA.x = w1p[k * Hdim];                  // B: K rows, N = n0+ln
            bA.y = w1p[k * Hdim + Hdim];
            bB.x = w1p[k * Hdim + 16];
            bB.y = w1p[k * Hdim + Hdim + 16];
            h0 = __builtin_amdgcn_wmma_f32_16x16x4_f32(false, a, false, bA,
                                                       (short)0, h0, false, false);
            h1 = __builtin_amdgcn_wmma_f32_16x16x4_f32(false, a, false, bB,
                                                       (short)0, h1, false, false);
        }
        {
            float bias0 = b1e[n0 + ln];
            float bias1 = b1e[n0 + 16 + ln];
            int c0 = wv * 32 + ln;
            #pragma unroll
            for (int j = 0; j < 8; ++j) {
                float v0 = h0[j] + bias0; v0 = v0 > 0.f ? v0 : 0.f;
                float v1 = h1[j] + bias1; v1 = v1 > 0.f ? v1 : 0.f;
                Hc[mbase + j][c0]      = v0;
                Hc[mbase + j][c0 + 16] = v1;
            }
        }
        __syncthreads();

        // ---- stage 2: Y[:, wv*64 .. +64) += Hc @ W2[chunk rows, :]
        const int n0y = wv * 64;
        const float* w2p  = W2e + (chunk * HC + khalf) * D + n0y + ln;
        const float* hrow = &Hc[ln][khalf];
        #pragma unroll 4
        for (int kk = 0; kk < HC; kk += 4) {
            v2f a = *(const v2f*)(hrow + kk);
            #pragma unroll
            for (int nt = 0; nt < 4; ++nt) {
                v2f b;
                b.x = w2p[kk * D + nt * 16];
                b.y = w2p[kk * D + D + nt * 16];
                accY[nt] = __builtin_amdgcn_wmma_f32_16x16x4_f32(false, a, false, b,
                                                                 (short)0, accY[nt],
                                                                 false, false);
            }
        }
        __syncthreads();   // protect Hc before next chunk overwrites it
    }

    // ---- epilogue: add b2, scatter rows to out[pair, :]
    #pragma unroll
    for (int nt = 0; nt < 4; ++nt) {
        int col = wv * 64 + nt * 16 + ln;
        float bias = b2e[col];
        #pragma unroll
        for (int j = 0; j < 8; ++j) {
            int p = sPair[mbase + j];
            if (p >= 0) out[(size_t)p * D + col] = accY[nt][j] + bias;
        }
    }
}

extern "C" void kernel_launch(void* const* d_in, const int* in_sizes, int n_in,
                              void* d_out, int out_size, void* d_ws, size_t ws_size,
                              hipStream_t stream) {
    const float* x   = (const float*)d_in[0];
    const int*   idx = (const int*)  d_in[1];
    const float* W1  = (const float*)d_in[2];
    const float* b1  = (const float*)d_in[3];
    const float* W2  = (const float*)d_in[4];
    const float* b2  = (const float*)d_in[5];
    float* out = (float*)d_out;
    int*   ws  = (int*)d_ws;

    k_zero <<<1, 64, 0, stream>>>(ws);
    k_count<<<TK / 256, 256, 0, stream>>>(idx, ws);
    k_scan <<<1, 32, 0, stream>>>(ws);
    k_fill <<<TK / 256, 256, 0, stream>>>(idx, ws);
    k_moe  <<<MAX_TILES, 256, 0, stream>>>(x, W1, b1, W2, b2, ws, out);
}